// GaussianStrokes_9131100471782
// MI455X (gfx1250) — compile-verified
//
#include <hip/hip_runtime.h>
#include <hip/hip_bf16.h>

typedef __attribute__((ext_vector_type(2))) float v2f;
typedef __attribute__((ext_vector_type(8))) float v8f;

#define F_DIM   512
#define G_DIM   256
#define RES     256
#define NCOLS   2048   // 512 centers + 512 invsig + 256 corr + 768 colors

// ---------------------------------------------------------------------------
// Kernel 1: all linear heads as one f32 WMMA GEMM.
//   raw[n*2048 + c] = xf[n] . Wrow(c) + bias(c)
// One wave (32 threads) per 16-column tile; 128 tiles.
// A = xf (16x4 frag per step), B = W.T slice (4x16 frag).
// D row m depends only on A row m, so padding rows 4..15 of A may hold
// arbitrary data (we clamp the row index with m&3 for address safety and
// never read D rows 4..15) -> branch-free inner loop.
// ---------------------------------------------------------------------------
__global__ __launch_bounds__(32)
void gs_heads_wmma(const float* __restrict__ x,
                   const float* __restrict__ Wc,  const float* __restrict__ bc,
                   const float* __restrict__ Ws,  const float* __restrict__ bs,
                   const float* __restrict__ Wr,  const float* __restrict__ br,
                   const float* __restrict__ Wcol,const float* __restrict__ bcol,
                   float* __restrict__ raw) {
    const int lane    = threadIdx.x;          // 0..31 (wave32)
    const int colbase = blockIdx.x * 16;
    const int m       = lane & 15;            // A row (batch) / B column in tile
    const int koff    = (lane >= 16) ? 2 : 0; // which K half this lane holds
    const int c       = colbase + m;          // global head column

    const float* wrow; float bias;
    if (c < 512)       { wrow = Wc   + c * F_DIM;          bias = bc[c];          }
    else if (c < 1024) { wrow = Ws   + (c - 512) * F_DIM;  bias = bs[c - 512];    }
    else if (c < 1280) { wrow = Wr   + (c - 1024) * F_DIM; bias = br[c - 1024];   }
    else               { wrow = Wcol + (c - 1280) * F_DIM; bias = bcol[c - 1280]; }

    // Rows 4..15 alias rows 0..3 (their D rows are discarded).
    const float* xrow = x + (m & 3) * F_DIM;

    v8f acc = {};
    #pragma unroll 4
    for (int k = 0; k < F_DIM; k += 4) {
        const float2 av = *(const float2*)(xrow + k + koff);
        const float2 bv = *(const float2*)(wrow + k + koff);
        v2f a; a[0] = av.x; a[1] = av.y;
        v2f b; b[0] = bv.x; b[1] = bv.y;
        acc = __builtin_amdgcn_wmma_f32_16x16x4_f32(
            /*neg_a=*/false, a, /*neg_b=*/false, b,
            /*c_mod=*/(short)0, acc, /*reuse_a=*/false, /*reuse_b=*/false);
    }

    if (lane < 16) {
        // D row mm lives in acc[mm] for lanes 0..15 (N = lane)
        #pragma unroll
        for (int mm = 0; mm < 4; ++mm)
            raw[mm * NCOLS + c] = acc[mm] + bias;
    }
}

// ---------------------------------------------------------------------------
// Kernel 2: transform raw head outputs into per-gaussian render params.
//   P[(n*256+g)*8] = { ix, cx*ix, iy, cy*iy, 2*rho, 2c0-1, 2c1-1, 2c2-1 }
// ---------------------------------------------------------------------------
__global__ __launch_bounds__(256)
void gs_pack_params(const float* __restrict__ raw, float* __restrict__ P) {
    const int t = blockIdx.x * blockDim.x + threadIdx.x;   // 0..1023
    const int n = t >> 8;
    const int g = t & 255;
    const float* r = raw + n * NCOLS;

    const float cx  = r[2 * g];
    const float cy  = r[2 * g + 1];
    const float rsx = r[512 + 2 * g];
    const float rsy = r[512 + 2 * g + 1];
    const float rho = r[1024 + g];
    const float c0  = r[1280 + 3 * g];
    const float c1  = r[1280 + 3 * g + 1];
    const float c2  = r[1280 + 3 * g + 2];

    const float L2E = 1.4426950408889634f;
    const float ix = 16.0f / (1.0f + __builtin_amdgcn_exp2f(-rsx * L2E)) + 1.0f;
    const float iy = 16.0f / (1.0f + __builtin_amdgcn_exp2f(-rsy * L2E)) + 1.0f;

    float* p = P + t * 8;
    p[0] = ix;        p[1] = cx * ix;
    p[2] = iy;        p[3] = cy * iy;
    p[4] = 2.0f * rho;
    p[5] = 2.0f * c0 - 1.0f;
    p[6] = 2.0f * c1 - 1.0f;
    p[7] = 2.0f * c2 - 1.0f;
}

// ---------------------------------------------------------------------------
// Kernel 3: rasterize. One block = one image n x 4 rows. Each thread renders
// a 1x4 pixel strip (shared sy per gaussian). Params staged in LDS, read as
// broadcast b128 loads.  dist = sx*(sx - 2*rho*sy) + sy^2 ;  w = exp2(-d*log2e)
// ---------------------------------------------------------------------------
__global__ __launch_bounds__(256)
void gs_render(const float* __restrict__ P, float* __restrict__ out) {
    __shared__ float4 lds[G_DIM * 2];   // 256 gaussians * 8 floats = 8 KB

    const int n      = blockIdx.x >> 6;
    const int rowblk = blockIdx.x & 63;
    const int t      = threadIdx.x;

    const float4* gp = (const float4*)(P + n * G_DIM * 8);
    lds[t]       = gp[t];
    lds[t + 256] = gp[t + 256];
    __syncthreads();

    const int row  = rowblk * 4 + (t >> 6);
    const int col0 = (t & 63) * 4;
    const float inv_res = 1.0f / (float)RES;
    const float v  = (float)row * inv_res;
    const float u0 = (float)col0 * inv_res;
    const float NL2E = -1.4426950408889634f;

    float a0[4] = {0,0,0,0}, a1[4] = {0,0,0,0}, a2[4] = {0,0,0,0}, aw[4] = {0,0,0,0};

    #pragma unroll 2
    for (int g = 0; g < G_DIM; ++g) {
        const float4 pa = lds[2 * g];
        const float4 pb = lds[2 * g + 1];
        const float ix = pa.x, cxix = pa.y, iy = pa.z, cyiy = pa.w;
        const float rho2 = pb.x, c0 = pb.y, c1 = pb.z, c2 = pb.w;

        const float sy  = fmaf(v, iy, -cyiy);
        const float sy2 = sy * sy;
        const float tt  = rho2 * sy;

        #pragma unroll
        for (int j = 0; j < 4; ++j) {
            const float u  = u0 + (float)j * inv_res;
            const float sx = fmaf(u, ix, -cxix);
            const float d  = fmaf(sx, sx - tt, sy2);
            const float w  = __builtin_amdgcn_exp2f(d * NL2E);
            a0[j] = fmaf(w, c0, a0[j]);
            a1[j] = fmaf(w, c1, a1[j]);
            a2[j] = fmaf(w, c2, a2[j]);
            aw[j] += w;
        }
    }

    const int HW = RES * RES;
    float* o = out + n * 3 * HW + row * RES + col0;
    #pragma unroll
    for (int j = 0; j < 4; ++j) {
        o[j]          = a0[j] / aw[j];
        o[HW + j]     = a1[j] / aw[j];
        o[2 * HW + j] = a2[j] / aw[j];
    }
}

// ---------------------------------------------------------------------------
extern "C" void kernel_launch(void* const* d_in, const int* in_sizes, int n_in,
                              void* d_out, int out_size, void* d_ws, size_t ws_size,
                              hipStream_t stream) {
    const float* x    = (const float*)d_in[0];
    const float* Wc   = (const float*)d_in[1];
    const float* bc   = (const float*)d_in[2];
    const float* Ws   = (const float*)d_in[3];
    const float* bs   = (const float*)d_in[4];
    const float* Wr   = (const float*)d_in[5];
    const float* br   = (const float*)d_in[6];
    const float* Wcol = (const float*)d_in[7];
    const float* bcol = (const float*)d_in[8];
    // d_in[9] (Wm), d_in[10] (bm): dead code in the reference forward.

    float* raw    = (float*)d_ws;                 // 4 * 2048 floats = 32 KB
    float* params = raw + 4 * NCOLS;              // 4 * 256 * 8 floats = 32 KB
    float* out    = (float*)d_out;                // [4, 3, 256, 256] f32

    gs_heads_wmma<<<NCOLS / 16, 32, 0, stream>>>(x, Wc, bc, Ws, bs, Wr, br,
                                                 Wcol, bcol, raw);
    gs_pack_params<<<4, 256, 0, stream>>>(raw, params);
    gs_render<<<4 * (RES / 4), 256, 0, stream>>>(params, out);
}